// ContextQueryAttention_19868518711751
// MI455X (gfx1250) — compile-verified
//
#include <hip/hip_runtime.h>
#include <math.h>

typedef float v2f __attribute__((ext_vector_type(2)));
typedef float v8f __attribute__((ext_vector_type(8)));

#define BB    64
#define CONT  400
#define QUES  50
#define DD    128

// ---------------------------------------------------------------------------
// K0: rc[b,i] = c[b,i,:].w_c ;  cq[b,j] = q[b,j,:].w_q   (one wave per row)
// ---------------------------------------------------------------------------
__global__ void cqa_rowdots(const float* __restrict__ c, const float* __restrict__ q,
                            const float* __restrict__ W, float* __restrict__ rc,
                            float* __restrict__ cq) {
    int wave = (blockIdx.x * blockDim.x + threadIdx.x) >> 5;
    int lane = threadIdx.x & 31;
    const int NRC = BB * CONT;
    const int NCQ = BB * QUES;
    if (wave >= NRC + NCQ) return;
    const float* row; const float* w; float* out;
    if (wave < NRC) { row = c + (size_t)wave * DD;            w = W;      out = rc + wave; }
    else { int r = wave - NRC; row = q + (size_t)r * DD;      w = W + DD; out = cq + r; }
    float s = 0.f;
    for (int d = lane; d < DD; d += 32) s += row[d] * w[d];
    for (int off = 16; off > 0; off >>= 1) s += __shfl_down(s, off, 32);
    if (lane == 0) *out = s;
}

// ---------------------------------------------------------------------------
// K1: S[b,i,j] = rc[b,i] + cq[b,j] + (c[b,i,:]*w_cq) . q[b,j,:]
//     einsum term via V_WMMA_F32_16X16X4_F32, one wave per 16x16 tile.
//     Branchless inner loop: out-of-range N lanes load a clamped (valid)
//     address and their garbage results are simply not stored.
// ---------------------------------------------------------------------------
__global__ void cqa_score(const float* __restrict__ c, const float* __restrict__ q,
                          const float* __restrict__ W, const float* __restrict__ rc,
                          const float* __restrict__ cq, float* __restrict__ S) {
    int b = blockIdx.x, mt = blockIdx.y, nt = blockIdx.z;
    int lane = threadIdx.x, half = lane >> 4, lm = lane & 15;
    int m0 = mt * 16, n0 = nt * 16;
    const float* wcq = W + 2 * DD;
    const float* cb  = c + (size_t)b * CONT * DD;
    const float* qb  = q + (size_t)b * QUES * DD;
    int m = m0 + lm;                         // A-matrix row for this lane
    int n = n0 + lm;                         // B-matrix col for this lane
    bool nvalid = (n < QUES);
    int nld = nvalid ? n : (QUES - 1);       // clamped: always in-bounds
    const float* arow = cb + (size_t)m * DD;
    const float* brow = qb + (size_t)nld * DD;
    v8f acc = {};
    for (int k0 = 0; k0 < DD; k0 += 4) {
        int k = k0 + 2 * half;
        v2f a, bm;
        a.x  = arow[k]     * wcq[k];
        a.y  = arow[k + 1] * wcq[k + 1];
        bm.x = brow[k];
        bm.y = brow[k + 1];
        acc = __builtin_amdgcn_wmma_f32_16x16x4_f32(false, a, false, bm,
                                                    (short)0, acc, false, false);
    }
    float cqv = cq[b * QUES + nld];
    float* Sb = S + (size_t)b * CONT * QUES;
    for (int v = 0; v < 8; ++v) {
        int mm = m0 + v + 8 * half;
        if (nvalid) Sb[(size_t)mm * QUES + n] = acc[v] + rc[b * CONT + mm] + cqv;
    }
}

// ---------------------------------------------------------------------------
// K2: S_q = masked softmax over axis 2 (j), one wave per (b,i) row of 50
// ---------------------------------------------------------------------------
__global__ void cqa_softmax_q(const float* __restrict__ S, const int* __restrict__ c_len,
                              const int* __restrict__ q_len, float* __restrict__ Sq) {
    int row = blockIdx.x;                 // b*CONT + i
    int b = row / CONT, i = row % CONT;
    int lane = threadIdx.x;
    int cl = c_len[b], ql = q_len[b];
    bool crow = (i < cl);
    const float* Sr = S + (size_t)row * QUES;
    int j0 = lane, j1 = lane + 32;
    float v0 = -3.0e38f, v1 = -3.0e38f, m0f = 0.f, m1f = 0.f;
    if (j0 < QUES) { bool mk = crow && (j0 < ql); m0f = mk ? 1.f : 0.f; v0 = mk ? Sr[j0] : -3.0e38f; }
    if (j1 < QUES) { bool mk = crow && (j1 < ql); m1f = mk ? 1.f : 0.f; v1 = mk ? Sr[j1] : -3.0e38f; }
    float mx = fmaxf(v0, v1);
    for (int off = 16; off > 0; off >>= 1) mx = fmaxf(mx, __shfl_xor(mx, off, 32));
    float e0 = (j0 < QUES) ? __expf(v0 - mx) : 0.f;
    float e1 = (j1 < QUES) ? __expf(v1 - mx) : 0.f;
    float sm = e0 + e1;
    for (int off = 16; off > 0; off >>= 1) sm += __shfl_xor(sm, off, 32);
    float inv = 1.0f / sm;
    float* Or = Sq + (size_t)row * QUES;
    if (j0 < QUES) Or[j0] = m0f * e0 * inv;
    if (j1 < QUES) Or[j1] = m1f * e1 * inv;
}

// ---------------------------------------------------------------------------
// K3: S_c = masked softmax over axis 1 (i, 400 elems), one 128-thr block per (b,j)
// ---------------------------------------------------------------------------
__global__ void cqa_softmax_c(const float* __restrict__ S, const int* __restrict__ c_len,
                              const int* __restrict__ q_len, float* __restrict__ Sc) {
    int col = blockIdx.x;                 // b*QUES + j
    int b = col / QUES, j = col % QUES;
    int tid = threadIdx.x;
    int cl = c_len[b], ql = q_len[b];
    bool qcol = (j < ql);
    const float* Sb = S + (size_t)b * CONT * QUES + j;
    __shared__ float red[4];
    float vals[4], mks[4];
    float mx = -3.0e38f;
    for (int t = 0; t < 4; ++t) {
        int i = tid + t * 128;
        bool inb = (i < CONT);
        bool mk = inb && qcol && (i < cl);
        float v = mk ? Sb[(size_t)i * QUES] : -3.0e38f;
        vals[t] = v; mks[t] = mk ? 1.f : 0.f;
        if (inb) mx = fmaxf(mx, v);
    }
    for (int off = 16; off > 0; off >>= 1) mx = fmaxf(mx, __shfl_xor(mx, off, 32));
    int wv = tid >> 5;
    if ((tid & 31) == 0) red[wv] = mx;
    __syncthreads();
    mx = fmaxf(fmaxf(red[0], red[1]), fmaxf(red[2], red[3]));
    __syncthreads();
    float es[4]; float sm = 0.f;
    for (int t = 0; t < 4; ++t) {
        int i = tid + t * 128;
        es[t] = (i < CONT) ? __expf(vals[t] - mx) : 0.f;
        sm += es[t];
    }
    for (int off = 16; off > 0; off >>= 1) sm += __shfl_xor(sm, off, 32);
    if ((tid & 31) == 0) red[wv] = sm;
    __syncthreads();
    sm = red[0] + red[1] + red[2] + red[3];
    float inv = 1.0f / sm;
    float* Ob = Sc + (size_t)b * CONT * QUES + j;
    for (int t = 0; t < 4; ++t) {
        int i = tid + t * 128;
        if (i < CONT) Ob[(size_t)i * QUES] = mks[t] * es[t] * inv;
    }
}

// ---------------------------------------------------------------------------
// K4: t[b] = S_c[b]^T (50x400) @ c[b] (400x128), WMMA f32, K-loop over CONT.
//     Branchless: rows j>=50 load a clamped address; results never stored.
// ---------------------------------------------------------------------------
__global__ void cqa_tmat(const float* __restrict__ Sc, const float* __restrict__ c,
                         float* __restrict__ t) {
    int b = blockIdx.x, mt = blockIdx.y, nt = blockIdx.z;
    int lane = threadIdx.x, half = lane >> 4, lm = lane & 15;
    int m0 = mt * 16, n0 = nt * 16;
    const float* Scb = Sc + (size_t)b * CONT * QUES;
    const float* cb  = c  + (size_t)b * CONT * DD;
    int jrow = m0 + lm;
    bool jvalid = (jrow < QUES);
    int jld = jvalid ? jrow : (QUES - 1);    // clamped: always in-bounds
    int n = n0 + lm;
    v8f acc = {};
    for (int k0 = 0; k0 < CONT; k0 += 4) {
        int k = k0 + 2 * half;
        v2f a, bm;
        a.x  = Scb[(size_t)k * QUES + jld];
        a.y  = Scb[(size_t)(k + 1) * QUES + jld];
        bm.x = cb[(size_t)k * DD + n];
        bm.y = cb[(size_t)(k + 1) * DD + n];
        acc = __builtin_amdgcn_wmma_f32_16x16x4_f32(false, a, false, bm,
                                                    (short)0, acc, false, false);
    }
    float* tb = t + (size_t)b * QUES * DD;
    for (int v = 0; v < 8; ++v) {
        int jj = m0 + v + 8 * half;
        if (jj < QUES) tb[(size_t)jj * DD + n] = acc[v];
    }
}

// ---------------------------------------------------------------------------
// K5: a = S_q@q, bmat = S_q@t (shared A operand, two accumulators), then
//     x = [c, a, c*a, c*bmat] written directly.
//     K = 50: 12 unguarded steps (k <= 47), then one masked tail step where
//     only the A operand is zeroed (0 * garbage-B == 0), addresses clamped.
// ---------------------------------------------------------------------------
__global__ void cqa_final(const float* __restrict__ Sq, const float* __restrict__ q,
                          const float* __restrict__ t, const float* __restrict__ c,
                          float* __restrict__ x) {
    int b = blockIdx.x, mt = blockIdx.y, nt = blockIdx.z;
    int lane = threadIdx.x, half = lane >> 4, lm = lane & 15;
    int m0 = mt * 16, n0 = nt * 16;
    const float* Sqb = Sq + (size_t)b * CONT * QUES;
    const float* qb  = q  + (size_t)b * QUES * DD;
    const float* tb  = t  + (size_t)b * QUES * DD;
    const float* cb  = c  + (size_t)b * CONT * DD;
    int m = m0 + lm;
    int n = n0 + lm;
    const float* arow = Sqb + (size_t)m * QUES;
    v8f accA = {}, accB = {};
    for (int k0 = 0; k0 < 48; k0 += 4) {     // k max = 44+2+1 = 47 < 50
        int k = k0 + 2 * half;
        v2f a, b1, b2;
        a.x  = arow[k];
        a.y  = arow[k + 1];
        b1.x = qb[(size_t)k * DD + n];
        b1.y = qb[(size_t)(k + 1) * DD + n];
        b2.x = tb[(size_t)k * DD + n];
        b2.y = tb[(size_t)(k + 1) * DD + n];
        accA = __builtin_amdgcn_wmma_f32_16x16x4_f32(false, a, false, b1,
                                                     (short)0, accA, false, false);
        accB = __builtin_amdgcn_wmma_f32_16x16x4_f32(false, a, false, b2,
                                                     (short)0, accB, false, false);
    }
    {   // tail step: k0 = 48 covers k = 48,49 (half 0) and 50,51 (half 1, masked)
        float mk = (half == 0) ? 1.f : 0.f;
        int kc = 48;                          // clamped row pair {48,49}: in-bounds
        v2f a, b1, b2;
        a.x  = arow[kc]     * mk;
        a.y  = arow[kc + 1] * mk;
        b1.x = qb[(size_t)kc * DD + n];
        b1.y = qb[(size_t)(kc + 1) * DD + n];
        b2.x = tb[(size_t)kc * DD + n];
        b2.y = tb[(size_t)(kc + 1) * DD + n];
        accA = __builtin_amdgcn_wmma_f32_16x16x4_f32(false, a, false, b1,
                                                     (short)0, accA, false, false);
        accB = __builtin_amdgcn_wmma_f32_16x16x4_f32(false, a, false, b2,
                                                     (short)0, accB, false, false);
    }
    for (int v = 0; v < 8; ++v) {
        int mm = m0 + v + 8 * half;
        float cv = cb[(size_t)mm * DD + n];
        float av = accA[v], bv = accB[v];
        float* xr = x + ((size_t)b * CONT + mm) * (4 * DD);
        xr[n]          = cv;
        xr[DD + n]     = av;
        xr[2 * DD + n] = cv * av;
        xr[3 * DD + n] = cv * bv;
    }
}

// ---------------------------------------------------------------------------
extern "C" void kernel_launch(void* const* d_in, const int* in_sizes, int n_in,
                              void* d_out, int out_size, void* d_ws, size_t ws_size,
                              hipStream_t stream) {
    const float* c     = (const float*)d_in[0];
    const float* q     = (const float*)d_in[1];
    const int*   c_len = (const int*)d_in[2];
    const int*   q_len = (const int*)d_in[3];
    const float* W     = (const float*)d_in[4];

    float* out = (float*)d_out;
    float* x   = out;                                       // B*CONT*4D
    float* Sq  = out + (size_t)BB * CONT * 4 * DD;          // B*CONT*QUES
    float* Sc  = Sq  + (size_t)BB * CONT * QUES;            // B*CONT*QUES

    float* ws = (float*)d_ws;
    float* S  = ws;                                         // B*CONT*QUES
    float* t  = S  + (size_t)BB * CONT * QUES;              // B*QUES*D
    float* rc = t  + (size_t)BB * QUES * DD;                // B*CONT
    float* cq = rc + (size_t)BB * CONT;                     // B*QUES

    int nrows = BB * CONT + BB * QUES;                      // 28800 waves
    cqa_rowdots<<<(nrows + 7) / 8, 256, 0, stream>>>(c, q, W, rc, cq);

    cqa_score<<<dim3(BB, CONT / 16, (QUES + 15) / 16), 32, 0, stream>>>(c, q, W, rc, cq, S);

    cqa_softmax_q<<<BB * CONT, 32, 0, stream>>>(S, c_len, q_len, Sq);
    cqa_softmax_c<<<BB * QUES, 128, 0, stream>>>(S, c_len, q_len, Sc);

    cqa_tmat<<<dim3(BB, (QUES + 15) / 16, DD / 16), 32, 0, stream>>>(Sc, c, t);

    cqa_final<<<dim3(BB, CONT / 16, DD / 16), 32, 0, stream>>>(Sq, q, t, c, x);
}